// EdgePredictor_11441792877014
// MI455X (gfx1250) — compile-verified
//
#include <hip/hip_runtime.h>

typedef __attribute__((ext_vector_type(2))) float v2f;
typedef __attribute__((ext_vector_type(8))) float v8f;

#define N_NODES 100000
#define N_EDGES 1000000
#define D_IN    128
#define D_HID   64

// ---------------- degree / normalization ----------------

__global__ void deg_init_kernel(float* __restrict__ deg, int n) {
    int i = blockIdx.x * blockDim.x + threadIdx.x;
    if (i < n) deg[i] = 1.0f;  // self loop contributes 1
}

__global__ void deg_count_kernel(const int* __restrict__ col, float* __restrict__ deg, int e) {
    int i = blockIdx.x * blockDim.x + threadIdx.x;
    if (i < e) atomicAdd(&deg[col[i]], 1.0f);
}

__global__ void deg_rsqrt_kernel(float* __restrict__ deg, int n) {
    int i = blockIdx.x * blockDim.x + threadIdx.x;
    if (i < n) deg[i] = rsqrtf(deg[i]);   // deg >= 1 always (self loop)
}

// ---------------- f32 WMMA GEMM: C[M x Nc] = A[M x K] @ B[K x Nc] ----------------
// Row-major. One wave per 16x16 C tile; blockDim.x = 128 (4 waves), wave w
// covers n-tile (blockIdx.y*4 + w). M tile = blockIdx.x. All dims multiples of 16.
//
// V_WMMA_F32_16X16X4_F32 VGPR layouts (ISA 7.12.2):
//   A 16x4 : lane l in [0,16): v0=A[l][k0+0], v1=A[l][k0+1]; lane 16+l: K+2/K+3
//   B 4x16 : lane l in [0,16): v0=B[k0+0][n0+l], v1=B[k0+1][n0+l]; lane 16+l: K+2/K+3
//   C 16x16: VGPR r: lanes 0-15 -> M=r, lanes 16-31 -> M=r+8; N = lane&15
__global__ void wmma_gemm_f32_kernel(const float* __restrict__ A,
                                     const float* __restrict__ B,
                                     float* __restrict__ C,
                                     int Nc, int K) {
    const int lane = threadIdx.x & 31;
    const int wave = threadIdx.x >> 5;
    const int m0   = blockIdx.x * 16;
    const int n0   = (blockIdx.y * 4 + wave) * 16;
    const int half = lane >> 4;   // 0: K offset {0,1}; 1: K offset {2,3}
    const int l    = lane & 15;

    v8f acc = {0.f, 0.f, 0.f, 0.f, 0.f, 0.f, 0.f, 0.f};

    const float* arow = A + (size_t)(m0 + l) * K;   // this lane's A row (M = l)
    const float* bcol = B + (n0 + l);               // this lane's B column (N = l)

    for (int k0 = 0; k0 < K; k0 += 4) {
        const int ka = k0 + 2 * half;
        v2f a, b;
        a[0] = arow[ka];
        a[1] = arow[ka + 1];
        b[0] = bcol[(size_t)ka * Nc];
        b[1] = bcol[(size_t)(ka + 1) * Nc];
        // (neg_a, A, neg_b, B, c_mod, C, reuse_a, reuse_b)
        acc = __builtin_amdgcn_wmma_f32_16x16x4_f32(false, a, false, b,
                                                    (short)0, acc, false, false);
    }

    float* crow = C + (size_t)(m0 + half * 8) * Nc + n0 + l;
#pragma unroll
    for (int r = 0; r < 8; ++r) {
        crow[(size_t)r * Nc] = acc[r];
    }
}

// ---------------- aggregation ----------------

// dst[i,c] = src[i,c] * dinv[i]^2 + bias[c]   (self-loop term + bias; also zero-inits dst)
__global__ void self_bias_kernel(const float* __restrict__ src,
                                 const float* __restrict__ dinv,
                                 const float* __restrict__ bias,
                                 float* __restrict__ dst,
                                 int n, int dshift) {
    int idx = blockIdx.x * blockDim.x + threadIdx.x;
    int d   = 1 << dshift;
    int i   = idx >> dshift;
    int c   = idx & (d - 1);
    if (i < n) {
        float w = dinv[i];
        w *= w;                       // = 1/deg[i]
        dst[idx] = src[idx] * w + bias[c];
    }
}

// dst[col[e], c] += src[row[e], c] * dinv[row]*dinv[col]
__global__ void scatter_kernel(const int* __restrict__ row,
                               const int* __restrict__ col,
                               const float* __restrict__ dinv,
                               const float* __restrict__ src,
                               float* __restrict__ dst,
                               int e, int dshift) {
    int idx = blockIdx.x * blockDim.x + threadIdx.x;
    int d   = 1 << dshift;
    int ei  = idx >> dshift;
    int c   = idx & (d - 1);
    if (ei < e) {
        int r  = row[ei];
        int cl = col[ei];
        float w = dinv[r] * dinv[cl];
        atomicAdd(&dst[((size_t)cl << dshift) + c],
                  src[((size_t)r << dshift) + c] * w);
    }
}

// ---------------- launch ----------------

extern "C" void kernel_launch(void* const* d_in, const int* in_sizes, int n_in,
                              void* d_out, int out_size, void* d_ws, size_t ws_size,
                              hipStream_t stream) {
    (void)in_sizes; (void)n_in; (void)out_size; (void)ws_size;

    const float* x  = (const float*)d_in[0];
    const int*   ei = (const int*)d_in[1];   // [2, E] flat: row then col
    const float* W1 = (const float*)d_in[2];
    const float* b1 = (const float*)d_in[3];
    const float* W2 = (const float*)d_in[4];
    const float* b2 = (const float*)d_in[5];
    const float* W3 = (const float*)d_in[6];
    const float* b3 = (const float*)d_in[7];

    float* outm = (float*)d_out;                       // m: N x 128
    float* outs = outm + (size_t)N_NODES * D_IN;       // s: N x 128

    const int* row = ei;
    const int* col = ei + N_EDGES;

    // workspace carve-out (all fit in MI455X's 192MB L2 per phase)
    float* ws   = (float*)d_ws;
    float* dinv = ws;  ws += ((N_NODES + 255) & ~255);          // 100k floats
    float* h1   = ws;  ws += (size_t)N_NODES * D_HID;           // x@W1
    float* h    = ws;  ws += (size_t)N_NODES * D_HID;           // layer-1 output
    float* t2   = ws;  ws += (size_t)N_NODES * D_IN;            // h@W2
    float* t3   = ws;                                           // h@W3

    const int thr = 256;

    // gcn_norm: deg (incl. self loops) -> deg^-1/2
    deg_init_kernel <<<(N_NODES + thr - 1) / thr, thr, 0, stream>>>(dinv, N_NODES);
    deg_count_kernel<<<(N_EDGES + thr - 1) / thr, thr, 0, stream>>>(col, dinv, N_EDGES);
    deg_rsqrt_kernel<<<(N_NODES + thr - 1) / thr, thr, 0, stream>>>(dinv, N_NODES);

    // layer 1: h1 = x @ W1 ; h = scatter(h1) + self + b1
    wmma_gemm_f32_kernel<<<dim3(N_NODES / 16, D_HID / 64), 128, 0, stream>>>(x, W1, h1, D_HID, D_IN);
    self_bias_kernel<<<(N_NODES * D_HID + thr - 1) / thr, thr, 0, stream>>>(h1, dinv, b1, h, N_NODES, 6);
    scatter_kernel  <<<(int)(((size_t)N_EDGES * D_HID + thr - 1) / thr), thr, 0, stream>>>(row, col, dinv, h1, h, N_EDGES, 6);

    // layers 2 & 3: t2 = h @ W2, t3 = h @ W3
    wmma_gemm_f32_kernel<<<dim3(N_NODES / 16, D_IN / 64), 128, 0, stream>>>(h, W2, t2, D_IN, D_HID);
    wmma_gemm_f32_kernel<<<dim3(N_NODES / 16, D_IN / 64), 128, 0, stream>>>(h, W3, t3, D_IN, D_HID);

    // aggregate directly into outputs
    self_bias_kernel<<<(N_NODES * D_IN + thr - 1) / thr, thr, 0, stream>>>(t2, dinv, b2, outm, N_NODES, 7);
    scatter_kernel  <<<(int)(((size_t)N_EDGES * D_IN + thr - 1) / thr), thr, 0, stream>>>(row, col, dinv, t2, outm, N_EDGES, 7);
    self_bias_kernel<<<(N_NODES * D_IN + thr - 1) / thr, thr, 0, stream>>>(t3, dinv, b3, outs, N_NODES, 7);
    scatter_kernel  <<<(int)(((size_t)N_EDGES * D_IN + thr - 1) / thr), thr, 0, stream>>>(row, col, dinv, t3, outs, N_EDGES, 7);
}